// HARLI_55430847922454
// MI455X (gfx1250) — compile-verified
//
#include <hip/hip_runtime.h>

typedef __attribute__((ext_vector_type(2))) float v2f;
typedef __attribute__((ext_vector_type(8))) float v8f;

#define LOG2E 1.4426950408889634f

__device__ __forceinline__ float fast_tanh(float x) {
    // tanh(x) = 1 - 2/(exp(2x)+1); exp2/rcp map to v_exp_f32 / v_rcp_f32
    float t = __builtin_amdgcn_exp2f(x * (2.0f * LOG2E));
    return 1.0f - 2.0f * __builtin_amdgcn_rcpf(t + 1.0f);
}
__device__ __forceinline__ float fast_sigmoid(float x) {
    return __builtin_amdgcn_rcpf(1.0f + __builtin_amdgcn_exp2f(-x * LOG2E));
}

// g0 = min(obs + (noise < hallucinogen), 1); zero the 98 Hebbian accumulators
__global__ __launch_bounds__(256) void ca_init(const float* __restrict__ obs,
                                               const float* __restrict__ noise,
                                               const float* __restrict__ hallu,
                                               float* __restrict__ g0,
                                               float* __restrict__ accum) {
    int i = blockIdx.x * 256 + threadIdx.x;   // 524288 total
    float h = hallu[0];
    float v = obs[i] + ((noise[i] < h) ? 1.0f : 0.0f);
    g0[i] = fminf(v, 1.0f);
    if (blockIdx.x == 0 && threadIdx.x < 98) accum[threadIdx.x] = 0.0f;
}

// One CA step. Block = 256 threads = 8 waves; each block owns a 16x16 tile of
// one image. Each wave handles 2 groups of 16 pixels via V_WMMA_F32_16X16X4_F32:
//   D(16 outs x 16 pixels) += A(16 outs x 4 chans) * B(4 chans x 16 pixels)
// 3 K-steps (K=9 padded to 12), 2 output tiles (32 channels) = 6 WMMAs / 16 px.
__global__ __launch_bounds__(256) void ca_step(const float* __restrict__ gin,
                                               float* __restrict__ gout,
                                               const float* __restrict__ w0,
                                               const float* __restrict__ w1,
                                               const float* __restrict__ bias1p,
                                               float* __restrict__ partials) {
    const int bid  = blockIdx.x;           // 2048 blocks
    const int n    = bid >> 8;             // image 0..7
    const int tile = bid & 255;
    const int ty   = (tile >> 4) << 4;
    const int tx   = (tile & 15) << 4;
    const int tid  = threadIdx.x;
    const int wid  = tid >> 5;
    const int lane = tid & 31;
    const int pcol = lane & 15;
    const bool lowHalf = (lane < 16);
    const int khalf = lowHalf ? 0 : 2;     // B/A frag K-pair for this lane half
    const int cbase = lowHalf ? 0 : 8;     // D-frag channel base for this half
    const float bias = bias1p[0];
    const float* gimg = gin + n * 65536;

    // A fragments (w0 tiles). ISA layout: lanes0-15 {K0,K1}, lanes16-31 {K2,K3}, M=lane&15
    v2f a[2][3];
#pragma unroll
    for (int t = 0; t < 2; t++)
#pragma unroll
        for (int s = 0; s < 3; s++) {
            int o  = t * 16 + pcol;
            int k0 = s * 4 + khalf;
            a[t][s].x = (k0     < 9) ? w0[o * 9 + k0]     : 0.0f;
            a[t][s].y = (k0 + 1 < 9) ? w0[o * 9 + k0 + 1] : 0.0f;
        }

    // w1 entries for the 16 channels this lane's D fragments hold
    float w1v[16];
#pragma unroll
    for (int j = 0; j < 8; j++) {
        w1v[j]     = w1[cbase + j];
        w1v[8 + j] = w1[16 + cbase + j];
    }

    float hacc[16];
#pragma unroll
    for (int j = 0; j < 16; j++) hacc[j] = 0.0f;
    float oacc = 0.0f, gacc = 0.0f;

#pragma unroll
    for (int it = 0; it < 2; it++) {
        const int y = ty + wid * 2 + it;
        const int x = tx + pcol;

        // B fragment: torus neighbors. channel k -> (dy,dx) = (k/3-1, k%3-1)
        v2f b[3];
#pragma unroll
        for (int s = 0; s < 3; s++) {
            int k0 = s * 4 + khalf;
            float bx = 0.0f, by = 0.0f;
            if (k0 < 9) {
                int dy = k0 / 3 - 1, dx = k0 % 3 - 1;
                bx = gimg[(((y + dy) & 255) << 8) + ((x + dx) & 255)];
            }
            if (k0 + 1 < 9) {
                int k1 = k0 + 1;
                int dy = k1 / 3 - 1, dx = k1 % 3 - 1;
                by = gimg[(((y + dy) & 255) << 8) + ((x + dx) & 255)];
            }
            b[s].x = bx; b[s].y = by;
        }

        v8f d0 = {0.f,0.f,0.f,0.f,0.f,0.f,0.f,0.f};
        v8f d1 = {0.f,0.f,0.f,0.f,0.f,0.f,0.f,0.f};
        d0 = __builtin_amdgcn_wmma_f32_16x16x4_f32(false, a[0][0], false, b[0], (short)0, d0, false, false);
        d0 = __builtin_amdgcn_wmma_f32_16x16x4_f32(false, a[0][1], false, b[1], (short)0, d0, false, false);
        d0 = __builtin_amdgcn_wmma_f32_16x16x4_f32(false, a[0][2], false, b[2], (short)0, d0, false, false);
        d1 = __builtin_amdgcn_wmma_f32_16x16x4_f32(false, a[1][0], false, b[0], (short)0, d1, false, false);
        d1 = __builtin_amdgcn_wmma_f32_16x16x4_f32(false, a[1][1], false, b[1], (short)0, d1, false, false);
        d1 = __builtin_amdgcn_wmma_f32_16x16x4_f32(false, a[1][2], false, b[2], (short)0, d1, false, false);

        float hv[16];
#pragma unroll
        for (int r = 0; r < 8; r++) {
            hv[r]     = fast_tanh(d0[r]);
            hv[8 + r] = fast_tanh(d1[r]);
        }

        float dot = 0.0f;
#pragma unroll
        for (int j = 0; j < 16; j++) { dot += w1v[j] * hv[j]; hacc[j] += hv[j]; }
        dot += __shfl_xor(dot, 16, 32);   // combine the two lane-halves of this pixel
        float ov = fast_sigmoid(dot - bias);
        if (lowHalf) {
            gout[n * 65536 + (y << 8) + x] = ov;
            oacc += ov;
            gacc += b[1].x;               // k=4 (center) = g_in at this pixel
        }
    }

    // deterministic half-wave reductions (masks <16 never cross lane halves)
#pragma unroll
    for (int m = 1; m < 16; m <<= 1) {
#pragma unroll
        for (int j = 0; j < 16; j++) hacc[j] += __shfl_xor(hacc[j], m, 32);
        oacc += __shfl_xor(oacc, m, 32);
        gacc += __shfl_xor(gacc, m, 32);
    }

    __shared__ float hpart[8][32];
    __shared__ float opart[8], gpart[8];
    if ((lane & 15) == 0) {
#pragma unroll
        for (int j = 0; j < 8; j++) {
            hpart[wid][cbase + j]      = hacc[j];
            hpart[wid][16 + cbase + j] = hacc[8 + j];
        }
        if (lowHalf) { opart[wid] = oacc; gpart[wid] = gacc; }
    }
    __syncthreads();
    if (tid < 32) {
        float s = 0.0f;
#pragma unroll
        for (int w = 0; w < 8; w++) s += hpart[w][tid];
        partials[bid * 34 + tid] = s;
    } else if (tid == 32) {
        float s = 0.0f;
        for (int w = 0; w < 8; w++) s += gpart[w];
        partials[bid * 34 + 32] = s;
    } else if (tid == 33) {
        float s = 0.0f;
        for (int w = 0; w < 8; w++) s += opart[w];
        partials[bid * 34 + 33] = s;
    }
}

// Reduce 256 block-partials per image into means, accumulate Hebbian state.
// accum: [0..31]=S0 (oioj0 col), [32..63]=OJ0 (=oi1), [64..95]=S1, [96]=OI0, [97]=OJ1
__global__ __launch_bounds__(288) void ca_reduce(const float* __restrict__ partials,
                                                 float* __restrict__ accum) {
    __shared__ float means[8][34];
    int t = threadIdx.x;
    if (t < 272) {
        int n = t / 34, j = t % 34;
        const float* p = partials + n * 256 * 34 + j;
        float s = 0.0f;
        for (int b = 0; b < 256; b++) s += p[b * 34];
        means[n][j] = s * (1.0f / 65536.0f);
    }
    __syncthreads();
    if (t < 32) {
        float s0 = 0.0f, oj0 = 0.0f, s1 = 0.0f;
#pragma unroll
        for (int n = 0; n < 8; n++) {
            float hm = means[n][t];
            s0  += hm * means[n][32];       // m01[n,t] * gmean[n]
            oj0 += hm;
            s1  += means[n][33] * hm;       // m11[n] * m01[n,t]
        }
        accum[t]      += s0;
        accum[32 + t] += oj0 * 0.125f;
        accum[64 + t] += s1;
    } else if (t == 32) {
        float gi = 0.0f, om = 0.0f;
        for (int n = 0; n < 8; n++) { gi += means[n][32]; om += means[n][33]; }
        accum[96] += gi * 0.125f;
        accum[97] += om * 0.125f;
    }
}

__global__ __launch_bounds__(256) void ca_action(const float* __restrict__ g,
                                                 float* __restrict__ out) {
    int i = blockIdx.x * 256 + threadIdx.x;  // 32768
    int n = i >> 12;
    int r = (i >> 6) & 63;
    int c = i & 63;
    float v = g[n * 65536 + ((96 + r) << 8) + (96 + c)];
    out[i] = (v > 0.5f) ? 1.0f : 0.0f;
}

__global__ __launch_bounds__(320) void ca_weights(
        const float* __restrict__ w0, const float* __restrict__ w1,
        const float* __restrict__ eta0, const float* __restrict__ a0,
        const float* __restrict__ b0, const float* __restrict__ c0,
        const float* __restrict__ d0,
        const float* __restrict__ eta1, const float* __restrict__ a1,
        const float* __restrict__ b1, const float* __restrict__ c1,
        const float* __restrict__ d1,
        const float* __restrict__ accum, float* __restrict__ out) {
    int t = threadIdx.x;
    if (t < 288) {
        int o = t / 9, i = t % 9;
        float oioj0 = accum[o];        // constant across i
        float oi0   = accum[96];       // constant across i
        float oj0   = accum[32 + o];
        out[32768 + t] = w0[t] + eta0[t] * (a0[t] * oioj0 + b0[i] * oi0 + c0[o] * oj0 + d0[t]);
    } else {
        int j = t - 288;
        float oioj1 = accum[64 + j];
        float oi1   = accum[32 + j];   // == oj0 accumulation
        float oj1   = accum[97];
        out[32768 + 288 + j] = w1[j] + eta1[j] * (a1[j] * oioj1 + b1[j] * oi1 + c1[0] * oj1 + d1[j]);
    }
}

extern "C" void kernel_launch(void* const* d_in, const int* in_sizes, int n_in,
                              void* d_out, int out_size, void* d_ws, size_t ws_size,
                              hipStream_t stream) {
    (void)in_sizes; (void)n_in; (void)out_size; (void)ws_size;
    const float* obs   = (const float*)d_in[0];
    const float* noise = (const float*)d_in[1];
    const float* hallu = (const float*)d_in[2];
    const float* bias1 = (const float*)d_in[3];
    const float* w0    = (const float*)d_in[4];
    const float* w1    = (const float*)d_in[5];
    const float* eta0  = (const float*)d_in[6];
    const float* a0    = (const float*)d_in[7];
    const float* b0    = (const float*)d_in[8];
    const float* c0    = (const float*)d_in[9];
    const float* d0    = (const float*)d_in[10];
    const float* eta1  = (const float*)d_in[11];
    const float* a1    = (const float*)d_in[12];
    const float* b1    = (const float*)d_in[13];
    const float* c1    = (const float*)d_in[14];
    const float* d1    = (const float*)d_in[15];
    float* out = (float*)d_out;

    float* g0       = (float*)d_ws;          // 524288 f32
    float* g1       = g0 + 8 * 256 * 256;    // 524288 f32
    float* partials = g1 + 8 * 256 * 256;    // 2048*34 f32
    float* accum    = partials + 2048 * 34;  // 98 f32

    ca_init<<<2048, 256, 0, stream>>>(obs, noise, hallu, g0, accum);
    float* gin = g0; float* gout = g1;
    for (int s = 0; s < 32; s++) {
        ca_step<<<2048, 256, 0, stream>>>(gin, gout, w0, w1, bias1, partials);
        ca_reduce<<<1, 288, 0, stream>>>(partials, accum);
        float* tmp = gin; gin = gout; gout = tmp;
    }
    ca_action<<<128, 256, 0, stream>>>(gin, out);
    ca_weights<<<1, 320, 0, stream>>>(w0, w1, eta0, a0, b0, c0, d0,
                                      eta1, a1, b1, c1, d1, accum, out);
}